// FishVectEncoder_927712936494
// MI455X (gfx1250) — compile-verified
//
#include <hip/hip_runtime.h>

typedef float v2f __attribute__((ext_vector_type(2)));
typedef float v8f __attribute__((ext_vector_type(8)));
typedef float f4  __attribute__((ext_vector_type(4)));

#define HW  4096   // 64*64
#define CCH 256
#define KCL 16
#define NB  2

// ds_swizzle group-of-32: offset[15]=0, xor_mask=[14:10], or_mask=[9:5]=0,
// and_mask=[4:0]=0x1f  ->  lane exchanges with (lane ^ mask); one DS op,
// no bounds-check scaffolding (all lanes active).
#define SWZ_XOR(v, mask) \
  __int_as_float(__builtin_amdgcn_ds_swizzle(__float_as_int(v), (((mask) << 10) | 0x1f)))

// ---------------------------------------------------------------------------
// Kernel A: soft assignment.
// tao_k(pixel) = x·V_k - 0.5*||V_k||^2   (||x||^2 cancels in softmax, sigma=1)
// GEMM [8192 x 256] * [256 x 16] via V_WMMA_F32_16X16X4_F32.
// One wave handles 16 consecutive pixels; D layout: VGPR j holds pixel j
// (lanes 0-15) / pixel j+8 (lanes 16-31), lane%16 = cluster. Softmax over
// clusters = xor-swizzle reduction over the 16-lane half group.
// ---------------------------------------------------------------------------
__global__ __launch_bounds__(256) void fv_softassign_kernel(
    const float* __restrict__ x, const float* __restrict__ V,
    float* __restrict__ sa_out)
{
  const int lane  = threadIdx.x & 31;
  const int wave  = threadIdx.x >> 5;
  const int gwave = blockIdx.x * 8 + wave;
  const int pixBase = gwave * 16;            // 16 pixels per wave, never crosses batch
  const int n    = pixBase >> 12;            // /4096
  const int hw0  = pixBase & (HW - 1);
  const int row  = lane & 15;                // cluster id for this lane's D column
  const int hi   = lane >> 4;

  const float* xb = x + (size_t)n * CCH * HW + hw0;
  const float* vr = V + row * CCH;

  // 0.5*||V_row||^2 (V is 16KB, cache resident)
  float hn = 0.f;
  #pragma unroll 8
  for (int c = 0; c < CCH; ++c) { float v = vr[c]; hn += v * v; }
  hn *= 0.5f;

  v8f acc = {};
#if __has_builtin(__builtin_amdgcn_wmma_f32_16x16x4_f32)
  #pragma unroll 4
  for (int c0 = 0; c0 < CCH; c0 += 4) {
    v2f a, b;
    // A 16x4 f32 layout: lanes0-15 v0=K0,v1=K1 ; lanes16-31 v0=K2,v1=K3
    a.x = xb[(size_t)(c0 + 2 * hi    ) * HW + row];
    a.y = xb[(size_t)(c0 + 2 * hi + 1) * HW + row];
    // B 4x16 f32 layout: v0 = rows K0(l0-15)/K1(l16-31), v1 = rows K2/K3
    b.x = vr[c0 + hi];
    b.y = vr[c0 + 2 + hi];
    acc = __builtin_amdgcn_wmma_f32_16x16x4_f32(
        /*neg_a=*/false, a, /*neg_b=*/false, b,
        /*c_mod=*/(short)0, acc, /*reuse_a=*/false, /*reuse_b=*/false);
  }
#else
  // VALU fallback producing the same D layout
  #pragma unroll
  for (int j = 0; j < 8; ++j) {
    float d = 0.f;
    for (int c = 0; c < CCH; ++c)
      d = __builtin_fmaf(xb[(size_t)c * HW + j + hi * 8], vr[c], d);
    acc[j] = d;
  }
#endif

  #pragma unroll
  for (int j = 0; j < 8; ++j) {
    float t = acc[j] - hn;
    float m = t;
    m = fmaxf(m, SWZ_XOR(m, 1));
    m = fmaxf(m, SWZ_XOR(m, 2));
    m = fmaxf(m, SWZ_XOR(m, 4));
    m = fmaxf(m, SWZ_XOR(m, 8));
    float e = __expf(t - m);
    float s = e;
    s += SWZ_XOR(s, 1);
    s += SWZ_XOR(s, 2);
    s += SWZ_XOR(s, 4);
    s += SWZ_XOR(s, 8);
    float w = e / s;
    const int hw = hw0 + j + hi * 8;         // pixel M = j + 8*hi
    sa_out[((size_t)n * KCL + row) * HW + hw] = w;
  }
}

// ---------------------------------------------------------------------------
// Kernel B: streaming residual encoder. Writes 269 MB -> the whole problem
// (~11.5 us at 23.3 TB/s). One thread: one (n,k,c) plane slice, 4 pixels.
// x & sa are L2-resident (8 MB + 0.5 MB, reused 16x/256x); outputs go out
// with NT stores so the write-once 269 MB does not evict the reused inputs.
// ---------------------------------------------------------------------------
__global__ __launch_bounds__(256) void fv_encode_kernel(
    const float* __restrict__ x, const float* __restrict__ V,
    const float* __restrict__ sa, float* __restrict__ enc)
{
  const int n  = blockIdx.z >> 4;
  const int k  = blockIdx.z & 15;
  const int c  = blockIdx.y;
  const int hw = (blockIdx.x * 256 + threadIdx.x) * 4;

  const float vkc = V[k * CCH + c];          // uniform -> scalar load
  const f4 x4 = *(const f4*)(x  + (size_t)(n * CCH + c) * HW + hw);
  const f4 s4 = *(const f4*)(sa + (size_t)(n * KCL + k) * HW + hw);

  const float MU = 4.0f;                     // 1/sqrt(PI_FV) = 1/sqrt(0.0625)
  const float SG = 2.8284271247461903f;      // 1/sqrt(2*PI_FV)

  f4 d  = x4 - vkc;                          // sigma = 1
  f4 mu = d * s4 * MU;
  f4 sg = (d * d - 1.0f) * s4 * SG;

  const size_t base = (size_t)n * (2 * KCL * CCH) * HW;
  float* pmu = enc + base + (size_t)(            k * CCH + c) * HW + hw;
  float* psg = enc + base + (size_t)(KCL * CCH + k * CCH + c) * HW + hw;
  __builtin_nontemporal_store(mu, (f4*)pmu);
  __builtin_nontemporal_store(sg, (f4*)psg);
}

extern "C" void kernel_launch(void* const* d_in, const int* in_sizes, int n_in,
                              void* d_out, int out_size, void* d_ws, size_t ws_size,
                              hipStream_t stream) {
  const float* x = (const float*)d_in[0];    // (2,256,64,64)
  const float* V = (const float*)d_in[1];    // (16,256)
  float* sa  = (float*)d_out;                          // (2,16,64,64)
  float* enc = (float*)d_out + (size_t)NB * KCL * HW;  // (2,8192,64,64)

  // A: 8192 pixels / 16 per wave = 512 waves = 64 blocks of 8 waves
  fv_softassign_kernel<<<64, 256, 0, stream>>>(x, V, sa);

  // B: grid (hw/1024, c, n*k), 256 thr * 4 px = 1024 px per block
  dim3 grid(HW / (256 * 4), CCH, NB * KCL);
  fv_encode_kernel<<<grid, 256, 0, stream>>>(x, V, sa, enc);
}